// ContinualGraph_36790689858197
// MI455X (gfx1250) — compile-verified
//
#include <hip/hip_runtime.h>

// Problem constants (from reference)
#define BATCH  2048
#define NCH    8
#define KSZ    256
#define SUBD   48
#define NHUBS  100000
#define FEAT   (NCH * SUBD)     // 384
#define BT     (BATCH / 32)     // 64 batch tiles of 32 rows
#define HT     64               // hub tiles (for parallel partial argmax)

typedef __attribute__((ext_vector_type(2))) float v2f;
typedef __attribute__((ext_vector_type(8))) float v8f;

// ---------------------------------------------------------------------------
// Phase 1: scores[b,c,k] = dot(x[b,c,:], codebooks[c,k,:]) via f32 WMMA.
// One wave computes one 16(M=batch) x 16(N=codebook entry) tile for one chunk.
// K = 48 -> 12 x V_WMMA_F32_16X16X4_F32 (full f32 precision; argmax over 100k
// near-ties is too precision-sensitive for bf16 inputs, and this GEMM is only
// ~0.4 GFLOP so its throughput is irrelevant).
// Output layout: scores[((bt*NCH + c)*KSZ + k)*32 + (row & 31)]  (bt = row/32)
// so phase 2 can stream one batch-tile's table contiguously into LDS.
// ---------------------------------------------------------------------------
__global__ void pq_scores_wmma(const float* __restrict__ x,
                               const float* __restrict__ cb,
                               const unsigned char* __restrict__ mask,
                               float* __restrict__ scores)
{
    const int tid  = threadIdx.x;
    const int lane = tid & 31;
    // global wave id: [c][mtile(128)][ntile(16)]
    const int gw   = blockIdx.x * 4 + (tid >> 5);
    const int c    = gw >> 11;          // / (128*16)
    const int rem  = gw & 2047;
    const int mt   = rem >> 4;          // batch tile of 16 rows
    const int nt   = rem & 15;          // codebook tile of 16 entries
    const int half = lane >> 4;         // 0: K pair {0,1}, 1: K pair {2,3}
    const int l16  = lane & 15;

    // A fragment source: row = mt*16 + l16 of x, chunk c
    const float* __restrict__ xa = x  + (size_t)(mt * 16 + l16) * FEAT + c * SUBD;
    // B fragment source: codebook entry = nt*16 + l16 of chunk c
    const float* __restrict__ bb = cb + (size_t)(c * KSZ + nt * 16 + l16) * SUBD;

    v8f acc = {};
#pragma unroll
    for (int kb = 0; kb < SUBD; kb += 4) {
        v2f a, b;
        // ISA 16x4 f32 A layout: VGPR0 = K={0|2}, VGPR1 = K={1|3} by lane half
        a[0] = xa[kb + 2 * half];
        a[1] = xa[kb + 2 * half + 1];
        // B (4x16) mirrors: N striped over lanes 0-15, K split by lane half
        b[0] = bb[kb + 2 * half];
        b[1] = bb[kb + 2 * half + 1];
        acc = __builtin_amdgcn_wmma_f32_16x16x4_f32(
            /*neg_a=*/false, a, /*neg_b=*/false, b,
            /*c_mod=*/(short)0, acc, /*reuse_a=*/false, /*reuse_b=*/false);
    }

    const float mv = mask[c] ? 1.0f : 0.0f;   // masked chunk contributes 0

    // D layout: VGPR j -> (M = j + 8*half, N = l16); consecutive j map to
    // consecutive (r & 31) slots -> compiler merges into global_store_b128 x2.
#pragma unroll
    for (int j = 0; j < 8; ++j) {
        const int r    = mt * 16 + j + half * 8;   // batch row
        const int kcol = nt * 16 + l16;            // codebook entry
        scores[((((size_t)(r >> 5)) * NCH + c) * KSZ + kcol) * 32 + (r & 31)] = acc[j] * mv;
    }
}

// ---------------------------------------------------------------------------
// Phase 2: partial argmax. Grid = (HT hub-tiles, BT batch-tiles), 256 threads.
// LDS holds the 32-row score table (8*256*32 f32 = 256 KB, dynamic LDS out of
// the 320 KB WGP pool). The fill uses the CDNA5 async Global->LDS path
// (global_load_async_to_lds_b128, tracked on ASYNCcnt) — no VGPR round-trip.
// lane = batch row within tile -> gathers hit 32 consecutive LDS dwords (no
// bank conflicts); wiring row is wave-uniform -> readfirstlane forces the
// scalar (s_load) path for the 8 indices.
// ---------------------------------------------------------------------------
__global__ void pq_argmax_partial(const float* __restrict__ scores,
                                  const int* __restrict__ wiring,
                                  float* __restrict__ pmax,
                                  int* __restrict__ pidx)
{
    extern __shared__ float tab[];               // NCH*KSZ*32 floats (dynamic)
    __shared__ float redM[8 * 32];
    __shared__ int   redI[8 * 32];

    const int tid = threadIdx.x;
    const int bt  = blockIdx.y;
    const int ht  = blockIdx.x;

    // Async-stream this batch-tile's table into LDS. Dynamic LDS begins at
    // byte offset groupstaticsize() (after redM/redI); the async instruction's
    // VDST operand is a per-lane LDS byte offset.
    {
        const unsigned lds_base = __builtin_amdgcn_groupstaticsize();
        const float4* __restrict__ src =
            (const float4*)(scores + (size_t)bt * (NCH * KSZ * 32));
        const int n4 = NCH * KSZ * 32 / 4;       // 16384 float4
        for (int i = tid; i < n4; i += 256) {
            const unsigned loff = lds_base + (unsigned)i * 16u;
            const float4* g = src + i;
            asm volatile("global_load_async_to_lds_b128 %0, %1, off"
                         :: "v"(loff), "v"(g) : "memory");
        }
        asm volatile("s_wait_asynccnt 0x0" ::: "memory");
    }
    __syncthreads();

    const int wave = tid >> 5;
    const int lane = tid & 31;

    const int per_ht = (NHUBS + HT - 1) / HT;
    int h0 = ht * per_ht;
    int h1 = h0 + per_ht; if (h1 > NHUBS) h1 = NHUBS;
    const int cnt   = h1 - h0;
    const int per_w = (cnt + 7) >> 3;
    int s = h0 + wave * per_w;
    int e = s + per_w; if (e > h1) e = h1;

    float best = -__builtin_inff();
    int   bidx = 0x7fffffff;
    for (int n = s; n < e; ++n) {
        const int nu = __builtin_amdgcn_readfirstlane(n);   // wave-uniform -> SGPR
        const int* __restrict__ wr = wiring + (size_t)nu * NCH;
        float en = tab[(0 * KSZ + wr[0]) * 32 + lane];
        en += tab[(1 * KSZ + wr[1]) * 32 + lane];
        en += tab[(2 * KSZ + wr[2]) * 32 + lane];
        en += tab[(3 * KSZ + wr[3]) * 32 + lane];
        en += tab[(4 * KSZ + wr[4]) * 32 + lane];
        en += tab[(5 * KSZ + wr[5]) * 32 + lane];
        en += tab[(6 * KSZ + wr[6]) * 32 + lane];
        en += tab[(7 * KSZ + wr[7]) * 32 + lane];
        if (en > best) { best = en; bidx = nu; }   // strict > keeps first max
    }

    redM[wave * 32 + lane] = best;
    redI[wave * 32 + lane] = bidx;
    __syncthreads();

    if (wave == 0) {
        float m  = redM[lane];
        int   ii = redI[lane];
#pragma unroll
        for (int w = 1; w < 8; ++w) {
            const float v = redM[w * 32 + lane];
            const int   j = redI[w * 32 + lane];
            if (v > m || (v == m && j < ii)) { m = v; ii = j; }
        }
        const int r = bt * 32 + lane;
        pmax[(size_t)r * HT + ht] = m;
        pidx[(size_t)r * HT + ht] = ii;
    }
}

// ---------------------------------------------------------------------------
// Phase 3: merge hub-tile partials per batch row; emit argmax index as float.
// ---------------------------------------------------------------------------
__global__ void pq_argmax_final(const float* __restrict__ pmax,
                                const int* __restrict__ pidx,
                                float* __restrict__ out)
{
    const int r = blockIdx.x * blockDim.x + threadIdx.x;
    if (r >= BATCH) return;
    float m  = pmax[(size_t)r * HT];
    int   ii = pidx[(size_t)r * HT];
    for (int h = 1; h < HT; ++h) {
        const float v = pmax[(size_t)r * HT + h];
        const int   j = pidx[(size_t)r * HT + h];
        if (v > m || (v == m && j < ii)) { m = v; ii = j; }
    }
    out[r] = (float)ii;
}

// ---------------------------------------------------------------------------
extern "C" void kernel_launch(void* const* d_in, const int* in_sizes, int n_in,
                              void* d_out, int out_size, void* d_ws, size_t ws_size,
                              hipStream_t stream)
{
    const float*         x      = (const float*)d_in[0];          // (2048, 384) f32
    const float*         cb     = (const float*)d_in[1];          // (8, 256, 48) f32
    const int*           wiring = (const int*)d_in[2];            // (100000, 8) int
    const unsigned char* mask   = (const unsigned char*)d_in[3];  // (8,) bool

    const size_t scores_bytes = (size_t)BATCH * NCH * KSZ * sizeof(float);   // 16 MB
    float* scores = (float*)d_ws;
    float* pmax   = (float*)((char*)d_ws + scores_bytes);
    int*   pidx   = (int*)  ((char*)d_ws + scores_bytes + (size_t)BATCH * HT * sizeof(float));

    // Phase 1: 8 chunks x 128 mtiles x 16 ntiles = 16384 waves; 4 waves/block
    pq_scores_wmma<<<16384 / 4, 128, 0, stream>>>(x, cb, mask, scores);

    // Phase 2: 256 KB dynamic LDS per block (320 KB/WGP available on CDNA5)
    pq_argmax_partial<<<dim3(HT, BT), 256, NCH * KSZ * 32 * sizeof(float), stream>>>(
        scores, wiring, pmax, pidx);

    // Phase 3
    pq_argmax_final<<<(BATCH + 255) / 256, 256, 0, stream>>>(pmax, pidx, (float*)d_out);
}